// DetNMSPostProcessor_71047349010503
// MI455X (gfx1250) — compile-verified
//
#include <hip/hip_runtime.h>
#include <stdint.h>

#define BATCHES 256
#define NQ      300
#define NC      80
#define TOPK    20
#define NT      256
#define SORTN   512
#define NW      10          // 10 x 32-bit words cover 300 suppression bits
#define IOU_THR_C   0.5f
#define SCORE_THR_C 0.6f

// ---- CDNA5 async DMA: global -> LDS, tracked by ASYNCcnt ----
__device__ __forceinline__ void async_load_b128_to_lds(uint32_t lds_addr, const void* gaddr) {
    asm volatile("global_load_async_to_lds_b128 %0, %1, off"
                 :: "v"(lds_addr), "v"(gaddr)
                 : "memory");
}

__device__ __forceinline__ void wait_async0() {
#if defined(__has_builtin)
#if __has_builtin(__builtin_amdgcn_s_wait_asynccnt)
    __builtin_amdgcn_s_wait_asynccnt(0);
#else
    asm volatile("s_wait_asynccnt 0" ::: "memory");
#endif
#else
    asm volatile("s_wait_asynccnt 0" ::: "memory");
#endif
}

__global__ void __launch_bounds__(NT)
detnms_kernel(const float* __restrict__ logits,
              const float* __restrict__ pboxes,
              const int*   __restrict__ sizes,
              float* __restrict__ out)
{
    __shared__ __align__(16) float ubox[NQ][4];   // async DMA target; cxcywh -> scaled xyxy in place
    __shared__ float    uscore[NQ];
    __shared__ int      ulabel[NQ];
    __shared__ float    skey[SORTN];
    __shared__ int      sidx[SORTN];
    __shared__ __align__(16) float sbox[NQ][4];   // sorted scaled xyxy (for output)
    __shared__ __align__(16) float obox[NQ][4];   // sorted class-offset boxes (for IoU)
    __shared__ float    sscore[NQ];
    __shared__ int      slabel[NQ];
    __shared__ float    sarea[NQ];
    __shared__ int      validf[NQ];
    __shared__ uint32_t supmask[NQ][NW];
    __shared__ uint32_t supfinal[NW];
    __shared__ int      sel[TOPK];
    __shared__ int      scnt;
    __shared__ int      maxbits;

    const int tid = threadIdx.x;
    const int b   = blockIdx.x;

    if (tid == 0) maxbits = 0;

    // ---- stage this image's boxes into LDS via async DMA (overlaps logit streaming) ----
    {
        uint32_t lds0 = (uint32_t)(uintptr_t)(&ubox[0][0]);
        const float* g = pboxes + (size_t)b * NQ * 4;
        int q = tid;                                     // 0..255 < 300
        async_load_b128_to_lds(lds0 + (uint32_t)q * 16u, g + (size_t)q * 4);
        q = tid + NT;
        if (q < NQ)
            async_load_b128_to_lds(lds0 + (uint32_t)q * 16u, g + (size_t)q * 4);
    }

    // ---- per-query class max/argmax + sigmoid (streams 96KB/image from HBM) ----
    const float* lrow_base = logits + (size_t)b * NQ * NC;
    for (int q = tid; q < NQ; q += NT) {
        const float* row = lrow_base + (size_t)q * NC;
        if (q + NT < NQ)
            __builtin_prefetch(lrow_base + (size_t)(q + NT) * NC, 0, 0);
        float best = -__builtin_inff();
        int   bl   = 0;
        for (int c = 0; c < NC; c += 4) {
            float4 v = *reinterpret_cast<const float4*>(row + c);
            if (v.x > best) { best = v.x; bl = c;     }
            if (v.y > best) { best = v.y; bl = c + 1; }
            if (v.z > best) { best = v.z; bl = c + 2; }
            if (v.w > best) { best = v.w; bl = c + 3; }
        }
        uscore[q] = 1.0f / (1.0f + __expf(-best));
        ulabel[q] = bl;
    }

    wait_async0();
    __syncthreads();

    // ---- cxcywh -> xyxy, scale by (s0,s1,s0,s1); track max |coord| for class offset ----
    const float sxf = (float)sizes[2 * b + 0];
    const float syf = (float)sizes[2 * b + 1];
    for (int q = tid; q < NQ; q += NT) {
        float cx = ubox[q][0], cy = ubox[q][1], w = ubox[q][2], h = ubox[q][3];
        float x1 = (cx - 0.5f * w) * sxf;
        float y1 = (cy - 0.5f * h) * syf;
        float x2 = (cx + 0.5f * w) * sxf;
        float y2 = (cy + 0.5f * h) * syf;
        ubox[q][0] = x1; ubox[q][1] = y1; ubox[q][2] = x2; ubox[q][3] = y2;
        float m = fmaxf(fmaxf(fabsf(x1), fabsf(y1)), fmaxf(fabsf(x2), fabsf(y2)));
        atomicMax(&maxbits, __float_as_int(m));   // abs values >= 0: int-bit compare is monotone
    }

    // ---- sort keys: masked scores, pad to 512 with -inf ----
    for (int i = tid; i < SORTN; i += NT) {
        float kv = -__builtin_inff();
        if (i < NQ) {
            float s = uscore[i];
            if (s > SCORE_THR_C) kv = s;
        }
        skey[i] = kv;
        sidx[i] = i;
    }
    __syncthreads();

    // ---- bitonic sort, descending (256 threads, 512 keys) ----
    for (int k = 2; k <= SORTN; k <<= 1) {
        for (int j = k >> 1; j > 0; j >>= 1) {
            int i = ((tid & ~(j - 1)) << 1) | (tid & (j - 1));
            int l = i | j;
            bool up = (i & k) == 0;
            float a = skey[i], c2 = skey[l];
            bool sw = up ? (a < c2) : (a > c2);
            if (sw) {
                skey[i] = c2; skey[l] = a;
                int t2 = sidx[i]; sidx[i] = sidx[l]; sidx[l] = t2;
            }
            __syncthreads();
        }
    }

    // ---- gather into sorted order; build offset boxes & areas ----
    const float max_coord = __int_as_float(maxbits) + 1.0f;
    for (int n = tid; n < NQ; n += NT) {
        float s  = skey[n];
        bool fin = s > -__builtin_inff();
        int  o   = sidx[n];
        if (o >= NQ) o = 0;                 // -inf padding entries; values unused
        int lb = ulabel[o];
        float x1, y1, x2, y2;
        if (fin) { x1 = ubox[o][0]; y1 = ubox[o][1]; x2 = ubox[o][2]; y2 = ubox[o][3]; }
        else     { x1 = y1 = x2 = y2 = 0.0f; lb = 0; }
        sscore[n] = s;
        slabel[n] = lb;
        sbox[n][0] = x1; sbox[n][1] = y1; sbox[n][2] = x2; sbox[n][3] = y2;
        float off = (float)lb * max_coord;
        obox[n][0] = x1 + off; obox[n][1] = y1 + off;
        obox[n][2] = x2 + off; obox[n][3] = y2 + off;
        sarea[n]  = (x2 - x1) * (y2 - y1);
        validf[n] = fin ? 1 : 0;
    }
    __syncthreads();

    // ---- suppression bitmask: bit j of row i set iff i (valid) would suppress j>i ----
    for (int t = tid; t < NQ * NW; t += NT) {
        int i = t / NW;
        int w = t - i * NW;
        uint32_t m = 0;
        if (validf[i]) {
            float ix1 = obox[i][0], iy1 = obox[i][1], ix2 = obox[i][2], iy2 = obox[i][3];
            float ia  = sarea[i];
            int jbase = w << 5;
            for (int bb = 0; bb < 32; ++bb) {
                int j = jbase + bb;
                if (j > i && j < NQ) {
                    float xx1 = fmaxf(ix1, obox[j][0]);
                    float yy1 = fmaxf(iy1, obox[j][1]);
                    float xx2 = fminf(ix2, obox[j][2]);
                    float yy2 = fminf(iy2, obox[j][3]);
                    float iw = fmaxf(xx2 - xx1, 0.0f);
                    float ih = fmaxf(yy2 - yy1, 0.0f);
                    float inter = iw * ih;
                    // iou > thr  <=>  inter > thr*(ia+ja-inter); division-free, false for degenerate 0-boxes
                    if (inter > IOU_THR_C * (ia + sarea[j] - inter)) m |= (1u << bb);
                }
            }
        }
        supmask[i][w] = m;
    }
    __syncthreads();

    // ---- greedy scan: one wave32, suppressed set lives in 10 lanes' registers ----
    if (tid < 32) {
        uint32_t sup = 0;                    // lane l owns bits [32l, 32l+32)
        for (int i = 0; i < NQ; ++i) {
            uint32_t wv = (uint32_t)__shfl((int)sup, i >> 5, 32);
            bool alive = ((wv >> (i & 31)) & 1u) == 0u;
            if (alive && tid < NW) sup |= supmask[i][tid];
        }
        if (tid < NW) supfinal[tid] = sup;
    }
    __syncthreads();

    // ---- compaction: first 20 kept in descending-score order ----
    if (tid == 0) {
        int cnt = 0;
        for (int n = 0; n < NQ; ++n) {
            bool kept = validf[n] && !((supfinal[n >> 5] >> (n & 31)) & 1u);
            if (kept && cnt < TOPK) sel[cnt++] = n;
        }
        scnt = cnt;
    }
    __syncthreads();

    // ---- write outputs: labels | boxes | scores | valid (all slots written) ----
    if (tid < TOPK) {
        int r = tid;
        float lab = -1.0f, sc = 0.0f, vv = 0.0f;
        float bx0 = 0.0f, bx1 = 0.0f, bx2 = 0.0f, bx3 = 0.0f;
        if (r < scnt) {
            int n = sel[r];
            lab = (float)slabel[n];
            sc  = sscore[n];
            bx0 = sbox[n][0]; bx1 = sbox[n][1]; bx2 = sbox[n][2]; bx3 = sbox[n][3];
            vv  = 1.0f;
        }
        float* out_labels = out;
        float* out_boxes  = out + (size_t)BATCHES * TOPK;
        float* out_scores = out + (size_t)BATCHES * TOPK * 5;
        float* out_valid  = out + (size_t)BATCHES * TOPK * 6;
        int s1 = b * TOPK + r;
        out_labels[s1]              = lab;
        out_boxes[(size_t)s1 * 4 + 0] = bx0;
        out_boxes[(size_t)s1 * 4 + 1] = bx1;
        out_boxes[(size_t)s1 * 4 + 2] = bx2;
        out_boxes[(size_t)s1 * 4 + 3] = bx3;
        out_scores[s1]              = sc;
        out_valid[s1]               = vv;
    }
}

extern "C" void kernel_launch(void* const* d_in, const int* in_sizes, int n_in,
                              void* d_out, int out_size, void* d_ws, size_t ws_size,
                              hipStream_t stream) {
    (void)in_sizes; (void)n_in; (void)d_ws; (void)ws_size; (void)out_size;
    const float* logits = (const float*)d_in[0];   // [256,300,80] f32
    const float* boxes  = (const float*)d_in[1];   // [256,300,4]  f32 cxcywh
    const int*   sizes  = (const int*)d_in[2];     // [256,2]      i32
    float* out = (float*)d_out;                    // 35840 f32
    detnms_kernel<<<BATCHES, NT, 0, stream>>>(logits, boxes, sizes, out);
}